// InteractiveGallingModelV5_65111704207990
// MI455X (gfx1250) — compile-verified
//
#include <hip/hip_runtime.h>
#include <hip/hip_bf16.h>

// ---------------------------------------------------------------------------
// InteractiveGallingModelV5 — Monte-Carlo scan, B=65536 rollouts x N=256 cycles
// Memory-bound streaming kernel (768 MB traffic, ~33 us floor @ 23.3 TB/s).
// One thread per rollout; carry (M, rho, mu) lives in registers for all cycles.
// Inputs staged via CDNA5 async-to-LDS (global_load_async_to_lds_b32) with an
// 8-stage, distance-7 prefetch pipeline gated by s_wait_asynccnt.
// Outputs written with non-temporal b32 stores (pure streaming).
// ---------------------------------------------------------------------------

#define MU_LOW   0.15f
#define MU_HIGH  1.0f
#define KDE_SM   0.005f
#define T_REF_C  165.0f

#define STAGES 8      // LDS pipeline stages (power of two)
#define PD     7      // prefetch distance (stages - 1)
#define BLK    256    // threads per block (8 wave32s)

// ---- fast transcendental helpers (single hardware op each) -----------------
__device__ __forceinline__ float fexp(float x) {           // e^x -> v_exp_f32
    return __builtin_amdgcn_exp2f(x * 1.44269504088896340736f);
}
__device__ __forceinline__ float frcp(float x) {           // 1/x -> v_rcp_f32
    return __builtin_amdgcn_rcpf(x);
}
__device__ __forceinline__ float fsigmoid(float x) {
    return frcp(1.0f + fexp(-x));
}
__device__ __forceinline__ float ftanh_pos(float x) {      // tanh, x >= 0
    float t = fexp(-2.0f * x);
    return (1.0f - t) * frcp(1.0f + t);
}
__device__ __forceinline__ float fpow(float a, float b) {  // a^b, a > 0
    return __builtin_amdgcn_exp2f(b * __builtin_amdgcn_logf(a));
}

// ---- CDNA5 async-to-LDS + wait --------------------------------------------
// Generic LDS pointers carry the LDS byte offset in their low 32 bits
// truncation yields the VDST byte address.
__device__ __forceinline__ unsigned lds_off(const void* p) {
    return (unsigned)(unsigned long long)p;
}
__device__ __forceinline__ void async_ld_b32(unsigned dst_lds, const float* src) {
    asm volatile("global_load_async_to_lds_b32 %0, %1, off"
                 :: "v"(dst_lds), "v"((unsigned long long)src)
                 : "memory");
}
__device__ __forceinline__ void wait_async_le_28() {
    // 8 stages x 4 streams in flight = 32; <=28 guarantees oldest stage done
    // (async loads complete in order within a wave).
    asm volatile("s_wait_asynccnt 0x1c" ::: "memory");
}

__global__ void __launch_bounds__(BLK)
galling_scan_kernel(const float* __restrict__ params,
                    const float* __restrict__ M_init,
                    const float* __restrict__ rho_init,
                    const float* __restrict__ T_in,
                    const float* __restrict__ ug,
                    const float* __restrict__ ud,
                    const float* __restrict__ us,
                    const float* __restrict__ nz,
                    float* __restrict__ out,
                    int B, int N)
{
    __shared__ float smem[STAGES][4][BLK];   // 32 KB, lane-private slots

    const int  tid = threadIdx.x;
    const long long b = (long long)blockIdx.x * BLK + tid;
    if (b >= B) return;

    // ---- uniform scalars (hoisted out of the scan, as in the reference) ----
    const float k_adh     = params[0];
    const float alpha_tau = params[1];
    const float p_g_base  = params[2];
    const float s_peak    = params[3];
    const float s_base    = params[4];
    const float decay     = params[5];
    const float p_d_base  = params[6];
    const float M_half    = params[7];
    const float c_stick   = params[8];
    const float f_min     = params[9];
    const float f_max     = params[10];
    const float M_sat     = params[11];
    const float rho_crit  = params[12];
    const float b_sharp   = params[13];

    const float dT        = T_in[0] - T_REF_C;
    const float softening = fexp(0.2f * dT);
    const float p_growth  = fminf(fmaxf(p_g_base * softening * frcp(1.0f + softening),
                                        0.01f), 0.99f);
    const float h_T       = fexp(-c_stick * dT);
    const float kg        = k_adh * softening;       // growth prefactor
    const float pdh       = p_d_base * h_T;          // detach prefactor
    const float inv_Msat  = frcp(M_sat);
    const float frange    = f_max - f_min;

    // ---- carry state -------------------------------------------------------
    float M   = M_init[b];
    float rho = rho_init[b];
    float mu  = MU_LOW;

    // ---- output row bases (rows: mu, rho, M, beta, p_growth, p_detach,
    //                               growth_events, detach_events) ------------
    const size_t RS = (size_t)N * (size_t)B;         // row stride
    float* o0 = out + 0 * RS + b;
    float* o1 = out + 1 * RS + b;
    float* o2 = out + 2 * RS + b;
    float* o3 = out + 3 * RS + b;
    float* o4 = out + 4 * RS + b;
    float* o5 = out + 5 * RS + b;
    float* o6 = out + 6 * RS + b;
    float* o7 = out + 7 * RS + b;

    // ---- async prefetch pipeline prologue: stages 0..PD-1 ------------------
    auto issue = [&](int slot, int tc) {
        const size_t g = (size_t)tc * (size_t)B + (size_t)b;
        async_ld_b32(lds_off(&smem[slot][0][tid]), ug + g);
        async_ld_b32(lds_off(&smem[slot][1][tid]), ud + g);
        async_ld_b32(lds_off(&smem[slot][2][tid]), us + g);
        async_ld_b32(lds_off(&smem[slot][3][tid]), nz + g);
    };
    #pragma unroll
    for (int s = 0; s < PD; ++s) {
        int tc = s < N ? s : (N - 1);
        issue(s, tc);
    }

    // ---- sequential scan over cycles --------------------------------------
    for (int t = 0; t < N; ++t) {
        // keep the pipeline full (clamp past the end; extra loads land in
        // cycle N-1's slot, rewriting it with the same data)
        int tp = t + PD; if (tp >= N) tp = N - 1;
        issue(tp & (STAGES - 1), tp);

        wait_async_le_28();                       // stage t resident in LDS

        const int sl = t & (STAGES - 1);
        const float ugv = smem[sl][0][tid];
        const float udv = smem[sl][1][tid];
        const float usv = smem[sl][2][tid];
        const float nzv = smem[sl][3][tid];

        // --- update_state ---
        const float beta   = fsigmoid(b_sharp * (rho - rho_crit));
        const bool  growth = ugv < p_growth;
        const float scale  = 4.0f * beta * (1.0f - beta) * s_peak + s_base;
        float dM = kg * fpow(mu, alpha_tau) * scale;
        dM = growth ? dM : 0.0f;

        const float g_M   = M * frcp(M + M_half + 1e-6f);
        const float p_det = fminf(fmaxf(pdh * g_M, 0.0f), 0.95f);
        const bool  det   = udv < p_det;
        const float sev   = f_min + usv * frange;

        float Mn = M + dM - decay * M;
        Mn = det ? Mn * (1.0f - sev) : Mn;
        Mn = fmaxf(Mn, 0.0f);

        const float rn = ftanh_pos(Mn * inv_Msat);

        // --- compute_friction ---
        const float bn  = fsigmoid(b_sharp * (rn - rho_crit));
        float mun = (1.0f - bn) * MU_LOW + bn * MU_HIGH;
        mun = fminf(fmaxf(mun + nzv * KDE_SM, 0.1f), 1.3f);

        // --- streaming (non-temporal) history writes ---
        const size_t tb = (size_t)t * (size_t)B;
        __builtin_nontemporal_store(mun,                  o0 + tb);
        __builtin_nontemporal_store(rn,                   o1 + tb);
        __builtin_nontemporal_store(Mn,                   o2 + tb);
        __builtin_nontemporal_store(bn,                   o3 + tb);
        __builtin_nontemporal_store(p_growth,             o4 + tb);
        __builtin_nontemporal_store(p_det,                o5 + tb);
        __builtin_nontemporal_store(growth ? 1.0f : 0.0f, o6 + tb);
        __builtin_nontemporal_store(det    ? 1.0f : 0.0f, o7 + tb);

        M = Mn; rho = rn; mu = mun;
    }
}

extern "C" void kernel_launch(void* const* d_in, const int* in_sizes, int n_in,
                              void* d_out, int out_size, void* d_ws, size_t ws_size,
                              hipStream_t stream) {
    const float* params   = (const float*)d_in[0];
    const float* M_init   = (const float*)d_in[1];
    const float* rho_init = (const float*)d_in[2];
    const float* T        = (const float*)d_in[3];
    const float* ug       = (const float*)d_in[4];
    const float* ud       = (const float*)d_in[5];
    const float* us       = (const float*)d_in[6];
    const float* nz       = (const float*)d_in[7];
    float*       out      = (float*)d_out;

    const int B = in_sizes[1];            // rollouts
    const int N = in_sizes[4] / B;        // cycles

    dim3 block(BLK);
    dim3 grid((B + BLK - 1) / BLK);       // 256 blocks x 8 wave32 = 2048 waves
    galling_scan_kernel<<<grid, block, 0, stream>>>(params, M_init, rho_init, T,
                                                    ug, ud, us, nz, out, B, N);
}